// TransitionUp_1400159339078
// MI455X (gfx1250) — compile-verified
//
#include <hip/hip_runtime.h>
#include <hip/hip_bf16.h>
#include <math.h>

typedef __attribute__((ext_vector_type(2))) float v2f;
typedef __attribute__((ext_vector_type(8))) float v8f;

#define BN_EPS 1e-5f
#define DIST_MIN 1e-16f

// Problem sizes (fixed by setup_inputs)
#define BB   4
#define NN   2048
#define MM   8192
#define CIN  256
#define COUT 128

// LDS row pitch padding: pitch = Kdim + 4 -> (pitch mod 64) == 4, so the 32
// lanes of a wave (16 N-rows x 2 K-halves, b64 each) hit disjoint bank pairs.
#define WPAD 4

// ---- CDNA5 async global->LDS staging (guarded; falls back to plain copy) ----
#if __has_builtin(__builtin_amdgcn_global_load_async_to_lds_b32) && \
    __has_builtin(__builtin_amdgcn_s_wait_asynccnt)
#define HAVE_ASYNC_LDS 1
#endif

typedef __attribute__((address_space(1))) int* gptr_i32;
typedef __attribute__((address_space(3))) int* lptr_i32;

__device__ __forceinline__ void lds_stage_b32(float* lds_dst, const float* gsrc) {
#ifdef HAVE_ASYNC_LDS
    __builtin_amdgcn_global_load_async_to_lds_b32(
        (gptr_i32)gsrc, (lptr_i32)lds_dst, /*offset=*/0, /*cpol=*/0);
#else
    *lds_dst = *gsrc;
#endif
}

__device__ __forceinline__ void lds_stage_wait() {
#ifdef HAVE_ASYNC_LDS
    __builtin_amdgcn_s_wait_asynccnt(0);
#endif
}

// ---------------------------------------------------------------------------
// Zero the BN stat accumulators (graph replays -> must re-init every call).
__global__ void init_stats_kernel(float* __restrict__ stats, int n) {
    int i = blockIdx.x * blockDim.x + threadIdx.x;
    if (i < n) stats[i] = 0.0f;
}

// ---------------------------------------------------------------------------
// H(rows x COUT) = X(rows x Kdim) * W(COUT x Kdim)^T via fp32 WMMA 16x16x4.
// One wave owns a 16-row x 128-col strip: A fragment loaded ONCE per K-step,
// reused across 8 column tiles (8x less A traffic, 8 WMMAs per A load).
// W is staged in LDS (async-to-LDS when available), bank-conflict padded.
// Also accumulates per-channel sum / sum-of-squares for BatchNorm.
__global__ void wmma_gemm_stats_kernel(const float* __restrict__ X,
                                       const float* __restrict__ W,
                                       float* __restrict__ H,
                                       float* __restrict__ stats, // [2*COUT]
                                       int rows, int Kdim, int kshift) {
    extern __shared__ float sW[];       // COUT * (Kdim + WPAD) floats
    const int tid   = threadIdx.x;
    const int pitch = Kdim + WPAD;

    // Stage W (COUT x Kdim, row-major) into padded LDS.
    const int wcount = COUT << kshift;  // COUT * Kdim, multiple of blockDim
    for (int i = tid; i < wcount; i += blockDim.x) {
        const int n = i >> kshift;
        const int k = i & (Kdim - 1);
        lds_stage_b32(&sW[n * pitch + k], &W[i]);
    }
    lds_stage_wait();
    __syncthreads();

    const int lane = tid & 31;
    const int wave = tid >> 5;
    const int row0 = (blockIdx.x * (blockDim.x >> 5) + wave) << 4;
    const int mn   = lane & 15;         // M (A) / N (B) index within a tile
    const int koff = (lane >> 4) << 1;  // K pair {0,1} vs {2,3}

    v8f acc[8];
    #pragma unroll
    for (int t = 0; t < 8; ++t) acc[t] = (v8f){};

    const float* __restrict__ xrow = X + (size_t)(row0 + mn) * Kdim;
    const float* __restrict__ brow = sW + mn * pitch + koff;

    for (int k = 0; k < Kdim; k += 4) {
        // 16x4 fp32 A fragment: lane holds (M=mn, K=k+koff .. +1)
        v2f a = *reinterpret_cast<const v2f*>(xrow + k + koff);
        __builtin_prefetch(xrow + k + 64, 0, 0);   // global_prefetch_b8
        #pragma unroll
        for (int t = 0; t < 8; ++t) {
            // 4x16 B fragment from LDS (B[k][n] = W[n][k]); ds_load_b64
            v2f b = *reinterpret_cast<const v2f*>(brow + (t << 4) * pitch + k);
            acc[t] = __builtin_amdgcn_wmma_f32_16x16x4_f32(
                /*neg_a=*/false, a, /*neg_b=*/false, b,
                /*c_mod=*/(short)0, acc[t], /*reuse_a=*/false, /*reuse_b=*/false);
        }
    }

    // C/D layout: VGPR j of lane l holds (M = j + 8*(l>>4), N = l&15).
    // All 8 values of a lane share one channel -> local reduce, one atomic.
    const int rbase = row0 + ((lane >> 4) << 3);
    #pragma unroll
    for (int t = 0; t < 8; ++t) {
        const int ncol = (t << 4) + mn;
        float s = 0.0f, s2 = 0.0f;
        #pragma unroll
        for (int j = 0; j < 8; ++j) {
            float v = acc[t][j];
            H[(size_t)(rbase + j) * COUT + ncol] = v;
            s  += v;
            s2 += v * v;
        }
        atomicAdd(&stats[ncol], s);
        atomicAdd(&stats[COUT + ncol], s2);
    }
}

// ---------------------------------------------------------------------------
// In-place BatchNorm(train) + ReLU using accumulated stats.
__global__ void normalize_relu_kernel(float* __restrict__ H,
                                      const float* __restrict__ stats,
                                      const float* __restrict__ gamma,
                                      const float* __restrict__ beta,
                                      int total, float invCount) {
    int i = blockIdx.x * blockDim.x + threadIdx.x;
    if (i >= total) return;
    int c = i & (COUT - 1);
    float mean = stats[c] * invCount;
    float var  = stats[COUT + c] * invCount - mean * mean;
    float rstd = rsqrtf(var + BN_EPS);
    float v = (H[i] - mean) * rstd * gamma[c] + beta[c];
    H[i] = fmaxf(v, 0.0f);
}

// ---------------------------------------------------------------------------
// kNN(k=3) per fine point; coarse positions staged in LDS (24 KB of 320 KB).
__global__ void knn_kernel(const float* __restrict__ p,      // (B, NN, 3) coarse
                           const float* __restrict__ p_old,  // (B, MM, 3) fine
                           float* __restrict__ wn,           // (B*MM, 3)
                           int*   __restrict__ idx) {        // (B*MM, 3)
    __shared__ float sp[NN * 3];
    const int b = blockIdx.y;
    const float* pb = p + (size_t)b * NN * 3;
    for (int i = threadIdx.x; i < NN * 3; i += blockDim.x)
        lds_stage_b32(&sp[i], &pb[i]);
    lds_stage_wait();
    __syncthreads();

    const int q = blockIdx.x * blockDim.x + threadIdx.x;
    if (q >= MM) return;
    const size_t qi = (size_t)b * MM + q;
    const float qx = p_old[qi * 3 + 0];
    const float qy = p_old[qi * 3 + 1];
    const float qz = p_old[qi * 3 + 2];

    float d0 = 3.0e38f, d1 = 3.0e38f, d2 = 3.0e38f;
    int   i0 = 0,       i1 = 0,       i2 = 0;
    for (int n = 0; n < NN; ++n) {
        float dx = qx - sp[n * 3 + 0];
        float dy = qy - sp[n * 3 + 1];
        float dz = qz - sp[n * 3 + 2];
        float d  = dx * dx + dy * dy + dz * dz;
        if (d < d2) {
            if (d < d1) {
                d2 = d1; i2 = i1;
                if (d < d0) { d1 = d0; i1 = i0; d0 = d; i0 = n; }
                else        { d1 = d;  i1 = n; }
            } else { d2 = d; i2 = n; }
        }
    }
    float w0 = 1.0f / fmaxf(d0, DIST_MIN);
    float w1 = 1.0f / fmaxf(d1, DIST_MIN);
    float w2 = 1.0f / fmaxf(d2, DIST_MIN);
    float inv = 1.0f / (w0 + w1 + w2);
    wn[qi * 3 + 0] = w0 * inv;
    wn[qi * 3 + 1] = w1 * inv;
    wn[qi * 3 + 2] = w2 * inv;
    idx[qi * 3 + 0] = i0;
    idx[qi * 3 + 1] = i1;
    idx[qi * 3 + 2] = i2;
}

// ---------------------------------------------------------------------------
// y = interp(gather of normalized x_up) + BN-ReLU(h_lat), fused; coalesced
// over channels (one block per fine point, one thread per channel).
__global__ void combine_kernel(const float* __restrict__ xup,   // (B*NN, C) normalized
                               const float* __restrict__ hlat,  // (B*MM, C) raw
                               const float* __restrict__ stats_lat,
                               const float* __restrict__ gamma,
                               const float* __restrict__ beta,
                               const float* __restrict__ wn,
                               const int*   __restrict__ idx,
                               float* __restrict__ y,
                               float invCount) {
    const int row = blockIdx.x;          // b*MM + m
    const int c   = threadIdx.x;         // channel
    const int b   = row / MM;
    const size_t base = (size_t)b * NN;

    const int   i0 = idx[row * 3 + 0];
    const int   i1 = idx[row * 3 + 1];
    const int   i2 = idx[row * 3 + 2];
    const float w0 = wn[row * 3 + 0];
    const float w1 = wn[row * 3 + 1];
    const float w2 = wn[row * 3 + 2];

    float interp = w0 * xup[(base + i0) * COUT + c]
                 + w1 * xup[(base + i1) * COUT + c]
                 + w2 * xup[(base + i2) * COUT + c];

    float mean = stats_lat[c] * invCount;
    float var  = stats_lat[COUT + c] * invCount - mean * mean;
    float rstd = rsqrtf(var + BN_EPS);
    float h    = hlat[(size_t)row * COUT + c];
    float lat  = fmaxf((h - mean) * rstd * gamma[c] + beta[c], 0.0f);

    y[(size_t)row * COUT + c] = interp + lat;
}

// ---------------------------------------------------------------------------
__global__ void copy_kernel(const float* __restrict__ src, float* __restrict__ dst, int n) {
    int i = blockIdx.x * blockDim.x + threadIdx.x;
    if (i < n) dst[i] = src[i];
}

// ---------------------------------------------------------------------------
extern "C" void kernel_launch(void* const* d_in, const int* in_sizes, int n_in,
                              void* d_out, int out_size, void* d_ws, size_t ws_size,
                              hipStream_t stream) {
    const float* x         = (const float*)d_in[0]; // (B, NN, CIN)
    const float* p         = (const float*)d_in[1]; // (B, NN, 3)
    const float* x_old     = (const float*)d_in[2]; // (B, MM, COUT)
    const float* p_old     = (const float*)d_in[3]; // (B, MM, 3)
    const float* W_up      = (const float*)d_in[4]; // (COUT, CIN)
    const float* gamma_up  = (const float*)d_in[5];
    const float* beta_up   = (const float*)d_in[6];
    const float* W_lat     = (const float*)d_in[7]; // (COUT, COUT)
    const float* gamma_lat = (const float*)d_in[8];
    const float* beta_lat  = (const float*)d_in[9];

    float* y_out   = (float*)d_out;                          // (B*MM, COUT)
    float* pos_out = (float*)d_out + (size_t)BB * MM * COUT; // (B*MM, 3)

    // Workspace carve-up (floats)
    float* ws        = (float*)d_ws;
    float* h_up      = ws;                                   // B*NN*COUT
    float* h_lat     = h_up + (size_t)BB * NN * COUT;        // B*MM*COUT
    float* stats_up  = h_lat + (size_t)BB * MM * COUT;       // 2*COUT
    float* stats_lat = stats_up + 2 * COUT;                  // 2*COUT
    float* wn        = stats_lat + 2 * COUT;                 // B*MM*3
    int*   knn_idx   = (int*)(wn + (size_t)BB * MM * 3);     // B*MM*3

    // 0) zero BN accumulators
    init_stats_kernel<<<1, 512, 0, stream>>>(stats_up, 4 * COUT);

    // 1) GEMM up: (B*NN x CIN) @ W_up^T -> h_up, + stats. 8 waves/block,
    //    each wave owns a 16x128 strip; W_up in LDS (128 rows x padded 260).
    {
        const int rows = BB * NN;                       // 8192
        const size_t shbytes = (size_t)COUT * (CIN + WPAD) * sizeof(float);
        wmma_gemm_stats_kernel<<<rows / 128, 256, shbytes, stream>>>(
            x, W_up, h_up, stats_up, rows, CIN, /*kshift=*/8);
    }

    // 2) GEMM lat: (B*MM x COUT) @ W_lat^T -> h_lat, + stats.
    {
        const int rows = BB * MM;                       // 32768
        const size_t shbytes = (size_t)COUT * (COUT + WPAD) * sizeof(float);
        wmma_gemm_stats_kernel<<<rows / 128, 256, shbytes, stream>>>(
            x_old, W_lat, h_lat, stats_lat, rows, COUT, /*kshift=*/7);
    }

    // 3) BN+ReLU on h_up (in place) -> normalized x_up for gathering
    {
        const int total = BB * NN * COUT;
        normalize_relu_kernel<<<total / 256, 256, 0, stream>>>(
            h_up, stats_up, gamma_up, beta_up, total, 1.0f / (BB * NN));
    }

    // 4) kNN(k=3), coarse positions in LDS
    {
        dim3 grid(MM / 256, BB);
        knn_kernel<<<grid, 256, 0, stream>>>(p, p_old, wn, knn_idx);
    }

    // 5) gather-interpolate + fused BN-ReLU(h_lat) + add -> y
    combine_kernel<<<BB * MM, COUT, 0, stream>>>(
        h_up, h_lat, stats_lat, gamma_lat, beta_lat, wn, knn_idx,
        y_out, 1.0f / (BB * MM));

    // 6) second tuple output: p_old passthrough
    {
        const int n = BB * MM * 3;
        copy_kernel<<<(n + 255) / 256, 256, 0, stream>>>(p_old, pos_out, n);
    }
}